// AttentionLite_35399120454010
// MI455X (gfx1250) — compile-verified
//
#include <hip/hip_runtime.h>
#include <hip/hip_bf16.h>
#include <math.h>

// Problem constants (match reference)
#define B_     2
#define C_     256
#define H_     48
#define W_     48
#define G_     2
#define HEADS_ 4
#define K_     7
#define PAD_   3
#define IN_W_  128
#define OUT_W_ 128
#define OW2_   64
#define HW_    (H_ * W_)          // 2304
#define HP_    (H_ + 2 * PAD_)    // 54
#define WP_    (W_ + 2 * PAD_)    // 54
#define NP_    (HP_ * WP_)        // 2916 padded pixels
#define NPIX_  HW_                // 2304 output pixels
#define J_     (G_ * K_ * K_)     // 98 attention positions
#define GH_    (G_ * HEADS_)      // 8 (group,head) rows
#define KPITCH_ 132               // padded LDS row pitch (528B, 16B-aligned,
                                  // 4-bank stride -> no 32-way conflicts)

typedef __attribute__((ext_vector_type(2))) float v2f;
typedef __attribute__((ext_vector_type(8))) float v8f;
typedef __attribute__((ext_vector_type(4))) int   v4i;

// ---------------------------------------------------------------------------
// CDNA5 async global->LDS staging (ASYNCcnt-tracked). The builtin takes a
// global (AS1) int4* source and an LDS (AS3) int4* destination plus imm
// offset / cpol. Guarded so the build stays green if a toolchain lacks it.
// ---------------------------------------------------------------------------
#if __has_builtin(__builtin_amdgcn_global_load_async_to_lds_b128) && \
    __has_builtin(__builtin_amdgcn_s_wait_asynccnt)
#define USE_ASYNC_LDS 1
typedef __attribute__((address_space(1))) v4i glob_v4i;
typedef __attribute__((address_space(3))) v4i lds_v4i;
__device__ __forceinline__ void async_copy16(void* lds_dst, const void* gsrc)
{
    __builtin_amdgcn_global_load_async_to_lds_b128(
        (glob_v4i*)(unsigned long long)gsrc,
        (lds_v4i*)(unsigned)(unsigned long long)lds_dst, 0, 0);
}
#else
#define USE_ASYNC_LDS 0
#endif

__device__ __forceinline__ void copy16(void* lds_dst, const void* gsrc)
{
#if USE_ASYNC_LDS
    async_copy16(lds_dst, gsrc);
#else
    *(float4*)lds_dst = *(const float4*)gsrc;
#endif
}

__device__ __forceinline__ void drain_async()
{
#if USE_ASYNC_LDS
    __builtin_amdgcn_s_wait_asynccnt(0);
#endif
}

// ---------------------------------------------------------------------------
// Kernel 1: k/v projection on the zero-padded 54x54 grid (blocked GEMM).
// Block = 8 waves. Each wave owns a 16-channel A-block, cached in 64 VGPRs
// for the whole block. The block walks 128 pixels in 8 subtiles; each
// subtile's B-panel (128K x 16N = 8KB) is staged once in LDS and shared by
// all 8 waves. K=128 in steps of 4 via v_wmma_f32_16x16x4_f32.
// blockIdx.y: 0 -> Wk/kbuf, 1 -> Wv/vbuf (128 channels each).
// Layout: kbuf/vbuf[b][g][padded_pixel][c]
// ---------------------------------------------------------------------------
__global__ __launch_bounds__(256) void AttnLite_kv_proj(
    const float* __restrict__ x, const float* __restrict__ Wk,
    const float* __restrict__ Wv, float* __restrict__ kbuf,
    float* __restrict__ vbuf)
{
    __shared__ float sB[IN_W_][16];   // [k][pixel] staged activations, 8KB

    const int t    = threadIdx.x;
    const int lane = t & 31;
    const int w    = t >> 5;          // wave id 0..7 -> channel subtile
    const int mn   = lane & 15;       // M (A row) == N (B col) slot
    const int hi   = lane >> 4;       // K pair {0,1} vs {2,3}
    const int pblk = blockIdx.x << 7; // 128-pixel block base (padded grid)
    const int och  = blockIdx.y;      // 0 = k, 1 = v
    const int b    = blockIdx.z >> 1;
    const int g    = blockIdx.z & 1;
    const int o0   = w << 4;          // channel base within 128-chunk

    // Cache this wave's A-block (16 x 128) in registers: 32 v2f per lane.
    const float* Wmat = (och == 0) ? Wk : Wv;
    const float* Wsrc = Wmat + (size_t)(g * OUT_W_ + o0 + mn) * IN_W_;
    v2f areg[32];
#pragma unroll
    for (int i = 0; i < 32; ++i)
        areg[i] = *(const v2f*)(Wsrc + 4 * i + 2 * hi);

    float* outbuf = (och == 0) ? kbuf : vbuf;

    // Staging role: fixed pixel column n_st, 8 channels stride 16.
    const int n_st = t & 15;
    const int k_st = t >> 4;

    for (int s = 0; s < 8; ++s) {
        const int p0 = pblk + (s << 4);
        {   // ---- stage B-panel into LDS (zero-padded borders) ----
            const int  p      = p0 + n_st;
            const bool pvalid = (p < NP_);
            const int  xp     = pvalid ? (p / WP_) : 0;
            const int  yp     = pvalid ? (p - xp * WP_) : 0;
            const int  xi     = xp - PAD_;
            const int  yi     = yp - PAD_;
            const bool inb    = pvalid && ((unsigned)xi < (unsigned)H_) &&
                                          ((unsigned)yi < (unsigned)W_);
            const float* src  = x + ((size_t)(b * C_ + g * IN_W_) * H_ +
                                     (inb ? xi : 0)) * W_ + (inb ? yi : 0);
#pragma unroll
            for (int it = 0; it < 8; ++it) {
                const int   k = k_st + (it << 4);
                const float v = src[(size_t)k * HW_];
                sB[k][n_st]   = inb ? v : 0.f;
            }
        }
        __syncthreads();

        // ---- 32 x v_wmma_f32_16x16x4_f32, A from registers, B from LDS ----
        v8f acc = {0.f, 0.f, 0.f, 0.f, 0.f, 0.f, 0.f, 0.f};
#pragma unroll
        for (int i = 0; i < 32; ++i) {
            v2f bv;
            bv.x = sB[4 * i + 2 * hi][mn];
            bv.y = sB[4 * i + 2 * hi + 1][mn];
            acc  = __builtin_amdgcn_wmma_f32_16x16x4_f32(
                false, areg[i], false, bv, (short)0, acc, false, false);
        }

        // ---- store 16x16 tile ----
        const int p = p0 + mn;
        if (p < NP_) {
            const size_t base = ((size_t)(b * G_ + g) * NP_ + p) * OUT_W_;
#pragma unroll
            for (int r = 0; r < 8; ++r)
                outbuf[base + (o0 + r + 8 * hi)] = acc[r];
        }
        __syncthreads();   // protect sB before next stage
    }
}

// ---------------------------------------------------------------------------
// Kernel 2: q projection (512 output channels, unpadded 48x48 grid),
// same blocked-GEMM structure. blockIdx.y = head (128 channels each).
// qbuf layout: [b][pixel][g][head][c]
// ---------------------------------------------------------------------------
__global__ __launch_bounds__(256) void AttnLite_q_proj(
    const float* __restrict__ x, const float* __restrict__ Wq,
    float* __restrict__ qbuf)
{
    __shared__ float sB[IN_W_][16];

    const int t    = threadIdx.x;
    const int lane = t & 31;
    const int w    = t >> 5;
    const int mn   = lane & 15;
    const int hi   = lane >> 4;
    const int pblk = blockIdx.x << 7;  // 18 blocks * 128 = 2304, exact
    const int head = blockIdx.y;       // 0..3
    const int b    = blockIdx.z >> 1;
    const int g    = blockIdx.z & 1;
    const int c0   = w << 4;           // channel base 0..112

    const float* Wsrc = Wq + (size_t)(g * (HEADS_ * OUT_W_) +
                                      head * OUT_W_ + c0 + mn) * IN_W_;
    v2f areg[32];
#pragma unroll
    for (int i = 0; i < 32; ++i)
        areg[i] = *(const v2f*)(Wsrc + 4 * i + 2 * hi);

    const int n_st = t & 15;
    const int k_st = t >> 4;

    for (int s = 0; s < 8; ++s) {
        const int p0 = pblk + (s << 4);
        {   // stage B (no padding on the q grid)
            const int    p   = p0 + n_st;
            const float* src = x + (size_t)(b * C_ + g * IN_W_) * HW_ + p;
#pragma unroll
            for (int it = 0; it < 8; ++it) {
                const int k = k_st + (it << 4);
                sB[k][n_st] = src[(size_t)k * HW_];
            }
        }
        __syncthreads();

        v8f acc = {0.f, 0.f, 0.f, 0.f, 0.f, 0.f, 0.f, 0.f};
#pragma unroll
        for (int i = 0; i < 32; ++i) {
            v2f bv;
            bv.x = sB[4 * i + 2 * hi][mn];
            bv.y = sB[4 * i + 2 * hi + 1][mn];
            acc  = __builtin_amdgcn_wmma_f32_16x16x4_f32(
                false, areg[i], false, bv, (short)0, acc, false, false);
        }

        const int p = p0 + mn;
#pragma unroll
        for (int r = 0; r < 8; ++r) {
            const int c = c0 + r + 8 * hi;
            qbuf[(((size_t)b * NPIX_ + p) * GH_ + g * HEADS_ + head) *
                     OUT_W_ + c] = acc[r];
        }
        __syncthreads();
    }
}

// ---------------------------------------------------------------------------
// Kernel 3: per-pixel attention. One 256-thread block per pixel.
// q (4KB) and the 98-row k-window (50KB, pitch-padded) are staged into LDS
// via async global->LDS loads, giving the 8x per-row reuse from LDS instead
// of cache. v streams from L2 (only 2x reuse, c-coalesced).
//   logits[gh][j] = dot128(q[gh], k[j]) + bias_h[gh][g'][kh] + bias_w[gh][g'][kw]
//   softmax over j (98); attn summed over heads before applying V.
// Output flat order (b, g, x, y, c) replicates the reference transpose+view.
// ---------------------------------------------------------------------------
__global__ __launch_bounds__(256) void AttnLite_attend(
    const float* __restrict__ qbuf, const float* __restrict__ kbuf,
    const float* __restrict__ vbuf, const float* __restrict__ rel_h,
    const float* __restrict__ rel_w, float* __restrict__ out)
{
    const int pix = blockIdx.x;   // 0..2303
    const int b   = blockIdx.y;
    const int xx  = pix / W_;
    const int yy  = pix - xx * W_;
    const int t   = threadIdx.x;

    __shared__ __attribute__((aligned(16))) float sQ[GH_ * OUT_W_]; // 1024
    __shared__ __attribute__((aligned(16))) float sK[J_ * KPITCH_]; // 12936
    __shared__ float sLog[GH_][J_];
    __shared__ float sBh[GH_][G_][K_];
    __shared__ float sBw[GH_][G_][K_];
    __shared__ float sAw[G_][J_];

    // ---- stage q: 256 x 16B chunks, one per thread ----
    const float* qsrc = qbuf + ((size_t)b * NPIX_ + pix) * (GH_ * OUT_W_);
    copy16(&sQ[t * 4], qsrc + t * 4);

    // ---- stage k-window: 98 rows x 32 chunks = 3136 chunk tasks ----
    for (int cid = t; cid < J_ * 32; cid += 256) {
        const int j  = cid >> 5;
        const int c4 = cid & 31;
        const int g2 = j / (K_ * K_);
        const int r  = j % (K_ * K_);
        const int kh = r / K_;
        const int kw = r % K_;
        const int kp = (xx + kh) * WP_ + (yy + kw);
        const float* src =
            kbuf + ((size_t)(b * G_ + g2) * NP_ + kp) * OUT_W_ + c4 * 4;
        copy16(&sK[j * KPITCH_ + c4 * 4], src);
    }
    drain_async();
    __syncthreads();

    // ---- separable rel-pos bias: 8(gh)*2(h/w)*2(g')*7(k) = 224 dot-64s ----
    if (t < GH_ * 2 * G_ * K_) {
        const int gh    = t / (2 * G_ * K_);
        int       rem   = t % (2 * G_ * K_);
        const int which = rem / (G_ * K_);   // 0 -> rel_h, 1 -> rel_w
        rem             = rem % (G_ * K_);
        const int g2    = rem / K_;
        const int kk    = rem % K_;
        const float* rel = which ? rel_w : rel_h;
        const float* qh  = sQ + gh * OUT_W_ + (which ? OW2_ : 0);
        float s = 0.f;
#pragma unroll 8
        for (int c = 0; c < OW2_; ++c)
            s += qh[c] * rel[((size_t)c * G_ + g2) * K_ + kk];
        if (which) sBw[gh][g2][kk] = s;
        else       sBh[gh][g2][kk] = s;
    }
    __syncthreads();

    // ---- logits: 8*98 = 784 dot-128s (LDS q x LDS k), 256 threads ----
    for (int l = t; l < GH_ * J_; l += 256) {
        const int gh = l / J_;
        const int j  = l % J_;
        const int g2 = j / (K_ * K_);
        const int r  = j % (K_ * K_);
        const int kh = r / K_;
        const int kw = r % K_;
        const float4* kv = (const float4*)(sK + j * KPITCH_);
        const float4* q  = (const float4*)(sQ + gh * OUT_W_);
        float s = 0.f;
#pragma unroll 8
        for (int c4 = 0; c4 < OUT_W_ / 4; ++c4) {
            const float4 kvv = kv[c4];
            const float4 qv  = q[c4];
            s += qv.x * kvv.x + qv.y * kvv.y + qv.z * kvv.z + qv.w * kvv.w;
        }
        sLog[gh][j] = s + sBh[gh][g2][kh] + sBw[gh][g2][kw];
    }
    __syncthreads();

    // ---- softmax over j for each of the 8 (g,head) rows ----
    if (t < GH_) {
        float mx = -3.0e38f;
        for (int j = 0; j < J_; ++j) mx = fmaxf(mx, sLog[t][j]);
        float sum = 0.f;
        for (int j = 0; j < J_; ++j) {
            const float e = __expf(sLog[t][j] - mx);
            sLog[t][j] = e;
            sum += e;
        }
        const float inv = 1.f / sum;
        for (int j = 0; j < J_; ++j) sLog[t][j] *= inv;
    }
    __syncthreads();

    // ---- head-reduced attention weights ----
    if (t < G_ * J_) {
        const int g = t / J_, j = t % J_;
        sAw[g][j] = sLog[g * HEADS_ + 0][j] + sLog[g * HEADS_ + 1][j] +
                    sLog[g * HEADS_ + 2][j] + sLog[g * HEADS_ + 3][j];
    }
    __syncthreads();

    // ---- out[g][c] = sum_j aw[g][j] * v[g'(j)][window(j)][c] ----
    const int g = t >> 7;
    const int c = t & 127;
    float acc = 0.f;
    for (int g2 = 0; g2 < G_; ++g2) {
        const float* vbase = vbuf + (size_t)(b * G_ + g2) * NP_ * OUT_W_;
        const float* aw    = &sAw[g][g2 * K_ * K_];
#pragma unroll
        for (int kh = 0; kh < K_; ++kh) {
            const int rowp = (xx + kh) * WP_ + yy;
#pragma unroll
            for (int kw = 0; kw < K_; ++kw) {
                acc += aw[kh * K_ + kw] *
                       vbase[(size_t)(rowp + kw) * OUT_W_ + c];
            }
        }
    }
    out[((size_t)(b * G_ + g) * NPIX_ + pix) * OUT_W_ + c] = acc;
}

// ---------------------------------------------------------------------------
// Host-side launch.  Inputs (setup_inputs order): x, Wq, Wk, Wv, rel_h, rel_w
// Workspace: qbuf (18.9 MB) | kbuf (6 MB) | vbuf (6 MB)  ~= 30.8 MB
// ---------------------------------------------------------------------------
extern "C" void kernel_launch(void* const* d_in, const int* in_sizes, int n_in,
                              void* d_out, int out_size, void* d_ws,
                              size_t ws_size, hipStream_t stream)
{
    (void)in_sizes; (void)n_in; (void)out_size; (void)ws_size;
    const float* x    = (const float*)d_in[0];
    const float* Wq   = (const float*)d_in[1];
    const float* Wk   = (const float*)d_in[2];
    const float* Wv   = (const float*)d_in[3];
    const float* relh = (const float*)d_in[4];
    const float* relw = (const float*)d_in[5];
    float*       out  = (float*)d_out;

    float* qbuf = (float*)d_ws;
    float* kbuf = qbuf + (size_t)B_ * NPIX_ * GH_ * OUT_W_;
    float* vbuf = kbuf + (size_t)B_ * G_ * NP_ * OUT_W_;

    // k/v projection: 23 x 128-pixel blocks, 2 x 128-channel chunks, (b,g)
    dim3 kvGrid((NP_ + 127) / 128, 2, B_ * G_);
    AttnLite_kv_proj<<<kvGrid, dim3(256, 1, 1), 0, stream>>>(x, Wk, Wv, kbuf,
                                                             vbuf);

    // q projection: 18 x 128-pixel blocks, 4 heads, (b,g)
    dim3 qGrid(NPIX_ / 128, HEADS_, B_ * G_);
    AttnLite_q_proj<<<qGrid, dim3(256, 1, 1), 0, stream>>>(x, Wq, qbuf);

    // attention: one 256-thread block per output pixel
    dim3 aGrid(NPIX_, B_, 1);
    AttnLite_attend<<<aGrid, dim3(256, 1, 1), 0, stream>>>(qbuf, kbuf, vbuf,
                                                           relh, relw, out);
}